// HungarianMatcher_40656160424039
// MI455X (gfx1250) — compile-verified
//
#include <hip/hip_runtime.h>

typedef float v2f __attribute__((ext_vector_type(2)));
typedef float v8f __attribute__((ext_vector_type(8)));

#define BSZ   16
#define NQ    1024
#define NTGT  96
#define NCLS  4
#define DD    6
#define NT_ALL (BSZ * NTGT)   /* 1536 */
#define NQ_ALL (BSZ * NQ)     /* 16384 */
#define HUNG_INF 1.0e9f

// ---------------------------------------------------------------------------
// Kernel 1: cost matrix. One wave per 16x16 (query x target) tile.
//   cost_class via V_WMMA_F32_16X16X4_F32:  probs(16x4) x (-onehot)(4x16)
//   cost_bbox  via VALU L1 over D=6, queries staged in LDS per workgroup.
// Also writes the per-batch diagonal block TRANSPOSED (t-major) to workspace
// so the Hungarian row scans are contiguous.
// ---------------------------------------------------------------------------
__global__ __launch_bounds__(128) void cost_kernel(
    const float* __restrict__ logits,   // (BSZ*NQ, NCLS)
    const float* __restrict__ boxes,    // (BSZ*NQ, DD)
    const int*   __restrict__ labels,   // (BSZ*NTGT)
    const float* __restrict__ tgt,      // (BSZ*NTGT, DD)
    float* __restrict__ C,              // (BSZ*NQ, NT_ALL)
    float* __restrict__ wsC,            // (BSZ, NTGT, NQ) transposed diag blocks
    int useWs)
{
    __shared__ float sprob[16][NCLS];
    __shared__ float sbox[16][DD];

    const int qbase = blockIdx.x * 16;
    const int tid   = threadIdx.x;

    // First 16 threads: softmax over NCLS=4 + stage query boxes in LDS.
    if (tid < 16) {
        const int q = qbase + tid;
        float l[NCLS];
#pragma unroll
        for (int c = 0; c < NCLS; ++c) l[c] = logits[q * NCLS + c];
        float mx = fmaxf(fmaxf(l[0], l[1]), fmaxf(l[2], l[3]));
        float s = 0.f;
#pragma unroll
        for (int c = 0; c < NCLS; ++c) { l[c] = expf(l[c] - mx); s += l[c]; }
        const float inv = 1.f / s;
#pragma unroll
        for (int c = 0; c < NCLS; ++c) sprob[tid][c] = l[c] * inv;
#pragma unroll
        for (int d = 0; d < DD; ++d) sbox[tid][d] = boxes[q * DD + d];
    }
    __syncthreads();

    const int wave  = tid >> 5;
    const int lane  = tid & 31;
    const int tbase = (blockIdx.y * 4 + wave) * 16;
    const int nsub  = lane & 15;          // row (A) / column (B,D) within tile
    const bool hi   = (lane >= 16);       // branch-free: keep EXEC all-ones for WMMA

    // A-matrix 16x4 f32 (ISA layout): vgpr0 = K0(lo)/K2(hi), vgpr1 = K1(lo)/K3(hi)
    v2f A;
    A.x = hi ? sprob[nsub][2] : sprob[nsub][0];
    A.y = hi ? sprob[nsub][3] : sprob[nsub][1];

    // B-matrix 4x16 f32 = -onehot(label): vgpr0 = K0(lo)/K2(hi), vgpr1 = K1(lo)/K3(hi)
    const int tl  = tbase + nsub;
    const int lab = labels[tl];
    v2f Bm;
    Bm.x = (lab == (hi ? 2 : 0)) ? -1.f : 0.f;
    Bm.y = (lab == (hi ? 3 : 1)) ? -1.f : 0.f;

    v8f acc = {};
    acc = __builtin_amdgcn_wmma_f32_16x16x4_f32(
        /*neg_a=*/false, A, /*neg_b=*/false, Bm,
        /*c_mod=*/(short)0, acc, /*reuse_a=*/false, /*reuse_b=*/false);

    // L1 bbox cost: this lane owns target tl; 8 query rows per lane half.
    float tb[DD];
#pragma unroll
    for (int d = 0; d < DD; ++d) tb[d] = tgt[tl * DD + d];

    const int qoff = hi ? 8 : 0;
    const int b    = qbase / NQ;
    const bool diag = (tbase >= b * NTGT) && (tbase < (b + 1) * NTGT);

#pragma unroll
    for (int r = 0; r < 8; ++r) {
        const int qi = qoff + r;               // query within tile (D-matrix row)
        float l1 = 0.f;
#pragma unroll
        for (int d = 0; d < DD; ++d) l1 += fabsf(sbox[qi][d] - tb[d]);
        const float val = acc[r] + l1;         // COST_BBOX = COST_CLASS = 1
        const long  qg  = (long)(qbase + qi);
        C[qg * (long)NT_ALL + tl] = val;
        if (useWs && diag) {
            wsC[((long)b * NTGT + (tl - b * NTGT)) * NQ + (qg - (long)b * NQ)] = val;
        }
    }
}

// ---------------------------------------------------------------------------
// Kernel 2: Jonker-Volgenant Hungarian, one wave32 per batch.
// LDS-resident dual/potential/state arrays; lanes parallelize the 1024-column
// scans; wave-level shfl_xor min+argmin (ties -> lowest j, matching argmin).
// Cost element (i0 in 1..NTGT, j in 1..NQ) = A[(i0-1)*rowStride + (j-1)*colStride].
// ---------------------------------------------------------------------------
__global__ __launch_bounds__(32) void hungarian_kernel(
    const float* __restrict__ A0, long batchStride, long rowStride, long colStride,
    int* __restrict__ outMatch)
{
    const int b = blockIdx.x;
    const float* A = A0 + (long)b * batchStride;

    __shared__ float v[NQ + 1];
    __shared__ float minv[NQ + 1];
    __shared__ float u[NTGT + 1];
    __shared__ int   p[NQ + 1];
    __shared__ int   way[NQ + 1];
    __shared__ int   used[NQ + 1];

    const int lane = threadIdx.x;

    for (int j = lane; j <= NQ; j += 32) { v[j] = 0.f; p[j] = 0; }
    for (int i = lane; i <= NTGT; i += 32) u[i] = 0.f;
    __syncthreads();

    for (int i = 1; i <= NTGT; ++i) {
        if (lane == 0) p[0] = i;
        for (int j = lane; j <= NQ; j += 32) { minv[j] = HUNG_INF; used[j] = 0; way[j] = 0; }
        __syncthreads();

        int j0 = 0;
        while (p[j0] != 0) {
            const int i0 = p[j0];
            if (lane == 0) used[j0] = 1;
            __syncthreads();

            const float ui0  = u[i0];
            const float* arow = A + (long)(i0 - 1) * rowStride;

            float best  = HUNG_INF;
            int   bestj = NQ + 1;
            for (int j = lane + 1; j <= NQ; j += 32) {
                if (!used[j]) {
                    float mv = minv[j];
                    const float cur = arow[(long)(j - 1) * colStride] - ui0 - v[j];
                    if (cur < mv) { mv = cur; minv[j] = cur; way[j] = j0; }
                    if (mv < best) { best = mv; bestj = j; }
                }
            }
            // wave32 min + argmin (tie-break: lowest column index)
#pragma unroll
            for (int off = 16; off > 0; off >>= 1) {
                const float ob = __shfl_xor(best, off, 32);
                const int   oj = __shfl_xor(bestj, off, 32);
                if (ob < best || (ob == best && oj < bestj)) { best = ob; bestj = oj; }
            }
            const float delta = best;
            __syncthreads();

            // used cols map to distinct rows -> conflict-free u[p[j]] updates
            for (int j = lane; j <= NQ; j += 32) {
                if (used[j]) { u[p[j]] += delta; v[j] -= delta; }
                else         { minv[j] -= delta; }
            }
            __syncthreads();
            j0 = bestj;
        }
        __syncthreads();

        if (lane == 0) {           // augmenting path walk (short, sequential)
            int jj = j0;
            while (jj != 0) { const int jn = way[jj]; p[jj] = p[jn]; jj = jn; }
        }
        __syncthreads();
    }

    // invert: matched query (0-based) for each target row
    for (int j = lane + 1; j <= NQ; j += 32) {
        const int r = p[j];
        if (r > 0) outMatch[b * NTGT + (r - 1)] = j - 1;
    }
}

// ---------------------------------------------------------------------------
extern "C" void kernel_launch(void* const* d_in, const int* in_sizes, int n_in,
                              void* d_out, int out_size, void* d_ws, size_t ws_size,
                              hipStream_t stream)
{
    (void)in_sizes; (void)n_in; (void)out_size;

    const float* logits = (const float*)d_in[0];   // (16,1024,4)  f32
    const float* boxes  = (const float*)d_in[1];   // (16,1024,6)  f32
    const int*   labels = (const int*)  d_in[2];   // (16,96)      i32
    const float* ctrl   = (const float*)d_in[3];   // (16,96,6)    f32

    float* C = (float*)d_out;                                   // 25,165,824 f32
    int*   match = (int*)(C + (size_t)NQ_ALL * NT_ALL);         // 1,536 i32 (bit-cast)

    float* wsC = (float*)d_ws;
    const size_t wsNeed = (size_t)BSZ * NTGT * NQ * sizeof(float);  // 6 MB
    const int useWs = (d_ws != nullptr && ws_size >= wsNeed) ? 1 : 0;

    dim3 grid(NQ_ALL / 16, NT_ALL / (16 * 4));   // (1024, 24), 4 waves/block
    cost_kernel<<<grid, 128, 0, stream>>>(logits, boxes, labels, ctrl, C, wsC, useWs);

    if (useWs) {
        hungarian_kernel<<<BSZ, 32, 0, stream>>>(
            wsC, (long)NTGT * NQ, (long)NQ, 1L, match);
    } else {
        // read diag blocks straight out of C (t is the fast axis there)
        hungarian_kernel<<<BSZ, 32, 0, stream>>>(
            C, (long)NQ * NT_ALL + NTGT, 1L, (long)NT_ALL, match);
    }
}